// LightGCN_4440996184480
// MI455X (gfx1250) — compile-verified
//
#include <hip/hip_runtime.h>

// LightGCN reference collapses: the graph is bipartite user->item, the GCN
// norm uses dis[src] with deg[src] structurally zero => norm == 0 for every
// edge => all 3 propagation layers are exactly zero and the edge-MLP/degree
// pipeline is dead. Bit-exact live computation:
//   final_user_h = user_emb * 0.25
//   final_item_h = (item_audio + artist_emb[aid] + album_emb[bid]) * 0.25
//   align_loss   = mean((final_item_h - item_audio @ W^T)^2)
// Everything is HBM-bound (~0.52 GB total => ~22us at 23.3 TB/s); the GEMM is
// done with V_WMMA_F32_16X16X4_F32 to stay on the matrix pipe at full f32.

#define NUM_USERS  500000
#define NUM_ITEMS  200000
#define EMBED_DIM  64

typedef __attribute__((ext_vector_type(2))) float v2f;
typedef __attribute__((ext_vector_type(8))) float v8f;

// ---------------------------------------------------------------- kernel 0
__global__ void init_loss_kernel(float* __restrict__ p) {
    if (threadIdx.x == 0 && blockIdx.x == 0) p[0] = 0.0f;
}

// ---------------------------------------------------------------- kernel 1
// final_user_h = user_emb * 0.25   (float4 streaming, pure HBM bandwidth)
__global__ void scale_users_kernel(const float4* __restrict__ u,
                                   float4* __restrict__ out, int n4) {
    int i = blockIdx.x * blockDim.x + threadIdx.x;
    if (i < n4) {
        float4 v = u[i];
        out[i] = make_float4(v.x * 0.25f, v.y * 0.25f, v.z * 0.25f, v.w * 0.25f);
    }
}

// ---------------------------------------------------------------- kernel 2
// final_item_h = (audio + artist_emb[aid] + album_emb[bid]) * 0.25
// One float4 (4 columns) per thread; gathers are 16B-aligned row slices.
__global__ void item_final_kernel(const float4* __restrict__ audio,
                                  const float4* __restrict__ artist,
                                  const float4* __restrict__ album,
                                  const int* __restrict__ artist_ids,
                                  const int* __restrict__ album_ids,
                                  float4* __restrict__ out, int n4) {
    int i = blockIdx.x * blockDim.x + threadIdx.x;
    if (i < n4) {
        int row = i >> 4;          // 16 float4 per 64-float row
        int c4  = i & 15;
        int aid = artist_ids[row];
        int bid = album_ids[row];
        float4 a = audio[i];
        float4 r = artist[aid * 16 + c4];
        float4 l = album[bid * 16 + c4];
        out[i] = make_float4((a.x + r.x + l.x) * 0.25f,
                             (a.y + r.y + l.y) * 0.25f,
                             (a.z + r.z + l.z) * 0.25f,
                             (a.w + r.w + l.w) * 0.25f);
    }
}

// ---------------------------------------------------------------- kernel 3
// projected = audio @ W^T via V_WMMA_F32_16X16X4_F32, fused with the
// squared-error reduction against final_item (already in d_out).
// One wave per 16-row tile; 8 waves (8 tiles) per 256-thread block.
// A (16x4): lane m=lane&15, VGPRs hold K = k0 + (lane<16 ? {0,1} : {2,3}).
// B (4x16): lane n=lane&15, same K split (B[k][n] = W[n][k], W staged in LDS).
// C/D (16x16): VGPR i holds row i (lanes 0-15) / row i+8 (lanes 16-31).
__global__ __launch_bounds__(256)
void proj_loss_kernel(const float* __restrict__ audio,
                      const float* __restrict__ W,
                      const float* __restrict__ final_item,
                      float* __restrict__ loss_acc) {
    __shared__ __align__(16) float Wl[EMBED_DIM * EMBED_DIM]; // 16 KB of 320 KB
    __shared__ float red[256];

    // Stage the 64x64 weight matrix into LDS once per block (reused by
    // 8 waves x 4 N-tiles x 16 K-steps).
    {
        const float4* ws = (const float4*)W;
        float4*       wd = (float4*)Wl;
        for (int i = threadIdx.x; i < (EMBED_DIM * EMBED_DIM) / 4; i += 256)
            wd[i] = ws[i];
    }
    __syncthreads();

    const int wave = threadIdx.x >> 5;
    const int lane = threadIdx.x & 31;
    const int tile = blockIdx.x * 8 + wave;   // 16-row tile index

    float lsum = 0.0f;
    if (tile < NUM_ITEMS / 16) {              // uniform per wave: EXEC all-ones
        const int m     = lane & 15;
        const int khalf = (lane < 16) ? 0 : 2;
        const float* arow = audio + (size_t)(tile * 16 + m) * EMBED_DIM;
        const float* brow = Wl + m * EMBED_DIM;   // n == lane&15 == m

        #pragma unroll
        for (int nt = 0; nt < 4; ++nt) {
            const int n0 = nt * 16;
            const float* bcol = brow + n0 * EMBED_DIM;
            v8f c = {};
            #pragma unroll
            for (int k0 = 0; k0 < EMBED_DIM; k0 += 4) {
                v2f a = *(const v2f*)(arow + k0 + khalf);
                v2f b = *(const v2f*)(bcol + k0 + khalf);
                // D = A x B + C, full f32 matrix pipe (16x16x4)
                c = __builtin_amdgcn_wmma_f32_16x16x4_f32(
                        /*neg_a=*/false, a, /*neg_b=*/false, b,
                        /*c_mod=*/(short)0, c,
                        /*reuse_a=*/false, /*reuse_b=*/false);
            }
            const int rbase = tile * 16 + ((lane < 16) ? 0 : 8);
            #pragma unroll
            for (int i = 0; i < 8; ++i) {
                float f = final_item[(size_t)(rbase + i) * EMBED_DIM + n0 + m];
                float d = f - c[i];
                lsum += d * d;
            }
        }
    }

    // Block reduction -> one atomicAdd per block.
    red[threadIdx.x] = lsum;
    __syncthreads();
    for (int s = 128; s > 0; s >>= 1) {
        if (threadIdx.x < s) red[threadIdx.x] += red[threadIdx.x + s];
        __syncthreads();
    }
    if (threadIdx.x == 0) atomicAdd(loss_acc, red[0]);
}

// ---------------------------------------------------------------- kernel 4
__global__ void finalize_loss_kernel(const float* __restrict__ acc,
                                     float* __restrict__ out) {
    if (threadIdx.x == 0 && blockIdx.x == 0)
        out[0] = acc[0] * (1.0f / (float)(NUM_ITEMS * EMBED_DIM));
}

// ----------------------------------------------------------------- launch
extern "C" void kernel_launch(void* const* d_in, const int* in_sizes, int n_in,
                              void* d_out, int out_size, void* d_ws, size_t ws_size,
                              hipStream_t stream) {
    const float* user_emb   = (const float*)d_in[0];
    const float* artist_emb = (const float*)d_in[1];
    const float* album_emb  = (const float*)d_in[2];
    const float* audio      = (const float*)d_in[3];
    const float* W          = (const float*)d_in[4];
    // d_in[5..11]: mlp_w1/b1/w2/b2, edge_attr, edge_weight_init, edge_index:
    // provably dead (norm == 0 structurally), never touched.
    const int* artist_ids   = (const int*)d_in[12];
    const int* album_ids    = (const int*)d_in[13];

    float* out       = (float*)d_out;
    float* out_user  = out;                                        // 500000*64
    float* out_item  = out + (size_t)NUM_USERS * EMBED_DIM;        // 200000*64
    float* out_loss  = out + (size_t)(NUM_USERS + NUM_ITEMS) * EMBED_DIM;
    float* loss_acc  = (float*)d_ws;

    init_loss_kernel<<<1, 32, 0, stream>>>(loss_acc);

    const int n4u = NUM_USERS * EMBED_DIM / 4;     // 8,000,000 float4
    scale_users_kernel<<<(n4u + 255) / 256, 256, 0, stream>>>(
        (const float4*)user_emb, (float4*)out_user, n4u);

    const int n4i = NUM_ITEMS * EMBED_DIM / 4;     // 3,200,000 float4
    item_final_kernel<<<(n4i + 255) / 256, 256, 0, stream>>>(
        (const float4*)audio, (const float4*)artist_emb, (const float4*)album_emb,
        artist_ids, album_ids, (float4*)out_item, n4i);

    const int ntiles  = NUM_ITEMS / 16;            // 12500
    const int nblocks = (ntiles + 7) / 8;          // 1563 (8 tiles per block)
    proj_loss_kernel<<<nblocks, 256, 0, stream>>>(audio, W, out_item, loss_acc);

    finalize_loss_kernel<<<1, 32, 0, stream>>>(loss_acc, out_loss);
}